// RGCNEncoder_85323820303220
// MI455X (gfx1250) — compile-verified
//
#include <hip/hip_runtime.h>

typedef __attribute__((ext_vector_type(2))) float v2f;
typedef __attribute__((ext_vector_type(8))) float v8f;

constexpr int N_NODES = 100000;
constexpr int N_REL   = 200;
constexpr int DIM     = 200;
constexpr int N_BASES = 30;
constexpr int N_EDGES = 800000;
constexpr int NCT     = (DIM + 15) / 16;          // 13 column tiles (208 cols, masked)
constexpr int MAXT    = N_EDGES / 16 + N_REL;     // upper bound on 16-edge tiles
constexpr int MSZ     = DIM * DIM;                // 40000

// Packed B-fragment layout: Mp[(k>>1)*2*DIM + 2*n + (k&1)] = M[k*DIM + n]
// -> lane's (W[kb][n], W[kb+1][n]) pair is one aligned 8-byte load.

// ---------------- small utility kernels ----------------
__global__ void zero_f32(float* p, int n) {
  int i = blockIdx.x * 256 + threadIdx.x;
  if (i < n) p[i] = 0.0f;
}
__global__ void zero_i32(int* p, int n) {
  int i = blockIdx.x * 256 + threadIdx.x;
  if (i < n) p[i] = 0;
}
__global__ void relu_k(float* p, int n) {
  int i = blockIdx.x * 256 + threadIdx.x;
  if (i < n) p[i] = fmaxf(p[i], 0.0f);
}

// count edges per relation + degree per (dst, relation)
__global__ void count_kernel(const int* __restrict__ etype, const int* __restrict__ dst,
                             int* __restrict__ cnt, float* __restrict__ deg) {
  int e = blockIdx.x * 256 + threadIdx.x;
  if (e >= N_EDGES) return;
  int r = etype[e];
  atomicAdd(&cnt[r], 1);
  atomicAdd(&deg[(long long)dst[e] * N_REL + r], 1.0f);
}

// serial scan over R=200 relations: edge offsets + tile offsets
__global__ void scan_kernel(const int* __restrict__ cnt, int* offr, int* cur,
                            int* tOff, int* nT) {
  if (threadIdx.x != 0 || blockIdx.x != 0) return;
  int run = 0, tr = 0;
  for (int r = 0; r < N_REL; ++r) {
    offr[r] = run; cur[r] = run; tOff[r] = tr;
    run += cnt[r];
    tr  += (cnt[r] + 15) >> 4;
  }
  nT[0] = tr;
}

// one thread per relation writes its tile metadata
__global__ void fill_tiles(const int* __restrict__ cnt, const int* __restrict__ offr,
                           const int* __restrict__ tOff, int* __restrict__ tRel,
                           int* __restrict__ tStart) {
  int r = blockIdx.x * 256 + threadIdx.x;
  if (r >= N_REL) return;
  int base = tOff[r], s = offr[r], c = cnt[r];
  int nt = (c + 15) >> 4;
  for (int i = 0; i < nt; ++i) { tRel[base + i] = r; tStart[base + i] = s + i * 16; }
}

__global__ void scatter_kernel(const int* __restrict__ etype, int* __restrict__ cur,
                               int* __restrict__ perm) {
  int e = blockIdx.x * 256 + threadIdx.x;
  if (e >= N_EDGES) return;
  int p = atomicAdd(&cur[etype[e]], 1);
  perm[p] = e;
}

__global__ void norm_kernel(const int* __restrict__ etype, const int* __restrict__ dst,
                            const float* __restrict__ deg, float* __restrict__ enorm) {
  int e = blockIdx.x * 256 + threadIdx.x;
  if (e >= N_EDGES) return;
  float d = deg[(long long)dst[e] * N_REL + etype[e]];
  enorm[e] = 1.0f / fmaxf(d, 1.0f);
}

// W[r] = sum_b comp[r,b] * bases[b], written directly in packed B-fragment order.
__global__ void build_w(const float* __restrict__ compL, const float* __restrict__ basesL,
                        float* __restrict__ Wp) {
  int i = blockIdx.x * 256 + threadIdx.x;     // i = k*DIM + n
  int r = blockIdx.y;
  if (i >= MSZ) return;
  int k = i / DIM, n = i - k * DIM;
  float s = 0.0f;
  for (int b = 0; b < N_BASES; ++b)
    s = fmaf(compL[r * N_BASES + b], basesL[(long long)b * MSZ + i], s);
  Wp[(long long)r * MSZ + (k >> 1) * (2 * DIM) + 2 * n + (k & 1)] = s;
}

// repack a plain [DIM,DIM] matrix (root) into packed B-fragment order
__global__ void pack_mat(const float* __restrict__ M, float* __restrict__ Mp) {
  int i = blockIdx.x * 256 + threadIdx.x;
  if (i >= MSZ) return;
  int k = i / DIM, n = i - k * DIM;
  Mp[(k >> 1) * (2 * DIM) + 2 * n + (k & 1)] = M[i];
}

// ---------------- WMMA GEMMs (fp32, 16x16x4) ----------------
// out[16 rows x DIM] = x[16 rows x DIM] @ M + bias   (M in packed fragment order)
__global__ __launch_bounds__(32) void root_gemm(const float* __restrict__ x,
                                                const float* __restrict__ Mp,
                                                const float* __restrict__ bias,
                                                float* __restrict__ out) {
  int t = blockIdx.x;                 // N_NODES/16 tiles, exact
  int lane = threadIdx.x;
  int m = lane & 15, kh = lane >> 4;  // row-in-tile, K-half
  const v2f* xp = (const v2f*)(x + (long long)(t * 16 + m) * DIM);
  v8f acc[NCT];
  v8f z8 = {0.f, 0.f, 0.f, 0.f, 0.f, 0.f, 0.f, 0.f};
#pragma unroll
  for (int ct = 0; ct < NCT; ++ct) acc[ct] = z8;
  for (int kk = 0; kk < DIM; kk += 4) {
    int kp = (kk >> 1) + kh;                    // K-pair index
    v2f a = xp[kp];
    const v2f* mp = (const v2f*)(Mp + kp * (2 * DIM));
#pragma unroll
    for (int ct = 0; ct < NCT; ++ct) {
      int n = ct * 16 + m;
      int nc = (n < DIM) ? n : (DIM - 1);       // clamp load, mask at store
      v2f b = mp[nc];
      acc[ct] = __builtin_amdgcn_wmma_f32_16x16x4_f32(false, a, false, b,
                                                      (short)0, acc[ct], false, false);
    }
  }
#pragma unroll
  for (int ct = 0; ct < NCT; ++ct) {
    int n = ct * 16 + m;
    if (n >= DIM) continue;
    float bv = bias[n];
#pragma unroll
    for (int v = 0; v < 8; ++v) {
      int row = v + kh * 8;
      out[(long long)(t * 16 + row) * DIM + n] = acc[ct][v] + bv;
    }
  }
}

// 16 edges (same relation) x DIM message GEMM, scaled by per-edge norm,
// scatter-added into out[dst].
__global__ __launch_bounds__(32) void edge_gemm(const float* __restrict__ x,
                                                const float* __restrict__ Wp,
                                                const int* __restrict__ src,
                                                const int* __restrict__ dst,
                                                const int* __restrict__ perm,
                                                const float* __restrict__ enorm,
                                                const int* __restrict__ tRel,
                                                const int* __restrict__ tStart,
                                                const int* __restrict__ offr,
                                                const int* __restrict__ cnt,
                                                const int* __restrict__ nT,
                                                float* __restrict__ out) {
  int t = blockIdx.x;
  if (t >= nT[0]) return;
  int lane = threadIdx.x;
  int r  = tRel[t];
  int tb = tStart[t];
  int relEnd = offr[r] + cnt[r];

  __shared__ int   s_src[16];
  __shared__ int   s_dst[16];
  __shared__ float s_nrm[16];
  if (lane < 16) {
    int s = tb + lane;
    bool ok = (s < relEnd);
    int e = perm[ok ? s : tb];       // tile is never empty; row 0 always valid
    s_src[lane] = src[e];
    s_dst[lane] = dst[e];
    s_nrm[lane] = ok ? enorm[e] : 0.0f;   // norm==0 masks invalid rows
  }
  __syncthreads();

  int m = lane & 15, kh = lane >> 4;
  const v2f* xp = (const v2f*)(x + (long long)s_src[m] * DIM);
  const float* Wr = Wp + (long long)r * MSZ;

  v8f acc[NCT];
  v8f z8 = {0.f, 0.f, 0.f, 0.f, 0.f, 0.f, 0.f, 0.f};
#pragma unroll
  for (int ct = 0; ct < NCT; ++ct) acc[ct] = z8;

  for (int kk = 0; kk < DIM; kk += 4) {
    int kp = (kk >> 1) + kh;
    v2f a = xp[kp];
    const v2f* wp = (const v2f*)(Wr + kp * (2 * DIM));
#pragma unroll
    for (int ct = 0; ct < NCT; ++ct) {
      int n = ct * 16 + m;
      int nc = (n < DIM) ? n : (DIM - 1);
      v2f b = wp[nc];
      acc[ct] = __builtin_amdgcn_wmma_f32_16x16x4_f32(false, a, false, b,
                                                      (short)0, acc[ct], false, false);
    }
  }

#pragma unroll
  for (int ct = 0; ct < NCT; ++ct) {
    int n = ct * 16 + m;
    if (n >= DIM) continue;
#pragma unroll
    for (int v = 0; v < 8; ++v) {
      int row = v + kh * 8;
      float nr = s_nrm[row];
      if (nr != 0.0f)
        atomicAdd(&out[(long long)s_dst[row] * DIM + n], acc[ct][v] * nr);
    }
  }
}

// ---------------- driver ----------------
extern "C" void kernel_launch(void* const* d_in, const int* in_sizes, int n_in,
                              void* d_out, int out_size, void* d_ws, size_t ws_size,
                              hipStream_t stream) {
  const int*   eidx  = (const int*)d_in[0];      // [2, E]
  const int*   src   = eidx;
  const int*   dst   = eidx + N_EDGES;
  const int*   etype = (const int*)d_in[1];      // [E]
  const float* emb   = (const float*)d_in[2];    // [N, DIM]
  const float* comp  = (const float*)d_in[3];    // [L, R, B]
  const float* bases = (const float*)d_in[4];    // [L, B, DIM, DIM]
  const float* root  = (const float*)d_in[5];    // [L, DIM, DIM]
  const float* bias  = (const float*)d_in[6];    // [L, DIM]
  float* out = (float*)d_out;                    // [N, DIM]

  // workspace layout (deg and the inter-layer activation alias: disjoint lifetimes)
  float* W     = (float*)d_ws;                               // R*MSZ packed (+pad)
  float* Rp    = W + ((long long)N_REL * MSZ + 64);          // packed root, MSZ (+pad)
  float* degx1 = Rp + (MSZ + 64);                            // max(N*R, N*DIM) = 20M
  int*   perm  = (int*)(degx1 + (long long)N_NODES * N_REL); // E
  float* enorm = (float*)(perm + N_EDGES);                   // E
  int*   cnt   = (int*)(enorm + N_EDGES);                    // R (padded slots)
  int*   offr  = cnt  + 256;
  int*   cur   = offr + 256;
  int*   tOff  = cur  + 256;
  int*   nT    = tOff + 256;
  int*   tRel  = nT   + 64;                                  // MAXT
  int*   tStart= tRel + MAXT;                                // MAXT
  float* deg = degx1;
  float* x1  = degx1;

  const int EB = (N_EDGES + 255) / 256;
  const int MB = (MSZ + 255) / 256;

  // ---- preprocessing: degrees, per-edge norm, relation-sorted edge permutation
  zero_f32<<<((long long)N_NODES * N_REL + 255) / 256, 256, 0, stream>>>(deg, N_NODES * N_REL);
  zero_i32<<<1, 256, 0, stream>>>(cnt, N_REL);
  count_kernel<<<EB, 256, 0, stream>>>(etype, dst, cnt, deg);
  scan_kernel<<<1, 1, 0, stream>>>(cnt, offr, cur, tOff, nT);
  fill_tiles<<<1, 256, 0, stream>>>(cnt, offr, tOff, tRel, tStart);
  scatter_kernel<<<EB, 256, 0, stream>>>(etype, cur, perm);
  norm_kernel<<<EB, 256, 0, stream>>>(etype, dst, deg, enorm);

  // ---- two RGCN layers
  for (int l = 0; l < 2; ++l) {
    const float* xin  = (l == 0) ? emb : x1;
    float*       xout = (l == 0) ? x1 : out;
    const float* compL  = comp  + (long long)l * N_REL * N_BASES;
    const float* basesL = bases + (long long)l * N_BASES * MSZ;
    const float* rootL  = root  + (long long)l * MSZ;
    const float* biasL  = bias  + (long long)l * DIM;

    build_w<<<dim3(MB, N_REL), 256, 0, stream>>>(compL, basesL, W);
    pack_mat<<<MB, 256, 0, stream>>>(rootL, Rp);
    root_gemm<<<N_NODES / 16, 32, 0, stream>>>(xin, Rp, biasL, xout);
    edge_gemm<<<MAXT, 32, 0, stream>>>(xin, W, src, dst, perm, enorm,
                                       tRel, tStart, offr, cnt, nT, xout);
    relu_k<<<((long long)N_NODES * DIM + 255) / 256, 256, 0, stream>>>(xout, N_NODES * DIM);
  }
}